// Patcher_19052474925281
// MI455X (gfx1250) — compile-verified
//
#include <hip/hip_runtime.h>
#include <hip/hip_bf16.h>
#include <stdint.h>

// 3D Haar DWT (reflect-pad unused for even dims) for x: [4,3,32,256,256] f32
// -> out: [4,24,16,128,128], subbands LLL,LLH,LHL,LHH,HLL,HLH,HHL,HHH
// (letters = temporal, vertical, horizontal). Net scale r^3 * 2*sqrt(2):
// applied as r (t-stage) * r (v-stage) * 2 (h-stage).
//
// MI455X plan: pure streaming (201 MB @ 23.3 TB/s ~ 8.6 us floor).
//  - 8 async global->LDS b128 DMAs per wave (4 KB in flight) before one
//    s_wait_asynccnt -> deep MLP to cover HBM latency.
//  - each thread computes all 8 subbands from its 2x2x2 blocks: every staged
//    float is read from LDS exactly once (no 8x read amplification).
//  - XOR-swizzled LDS slots -> 2-way (minimal) bank conflicts on b128 reads.
//  - non-temporal float4 stores, 512 B contiguous per wave per subband plane.

typedef float v4f __attribute__((ext_vector_type(4)));

namespace {
constexpr int Cc = 3;
constexpr int T  = 32;
constexpr int H  = 256;
constexpr int W  = 256;
constexpr int T2 = 16;
constexpr int H2 = 128;
constexpr int W2 = 128;
}

__device__ __forceinline__ int lds_slot(int c) {
    // permute 16B-chunk slot within a 1KB row: 2-way-minimal bank conflicts
    return c ^ (((c >> 4) & 3) << 2);
}

__global__ __launch_bounds__(256)
void haar3d_dwt_kernel(const float* __restrict__ x, float* __restrict__ out) {
    // 32 input rows (2 frames x 16 h-rows) x 256 floats = 32 KB staged in LDS
    __shared__ v4f sm[32 * 64];

    const int hb  = blockIdx.x;          // 0..15 : output rows 8*hb .. 8*hb+7
    const int t   = blockIdx.y;          // 0..15 : output frame
    const int bc  = blockIdx.z;          // b*3 + c
    const int tid = threadIdx.x;

    // ------------------------------------------------------------------
    // Stage 1: async DMA, 8 x 16B per lane = 4KB per wave in flight.
    // chunkid = tid + 256*i ; r = chunkid>>6 in 0..31 ; c = tid&63.
    // Row r: dt = r>>4 (frame), hh = r&15 (input h row within tile).
    // Each async instruction moves one contiguous 512B row segment.
    // ------------------------------------------------------------------
    {
        const int c = tid & 63;                    // 16B chunk within row
        const uint32_t smBase = (uint32_t)(uintptr_t)(&sm[0]);
        const uint32_t ldsCol = (uint32_t)(lds_slot(c) * 16);
#pragma unroll
        for (int i = 0; i < 8; ++i) {
            const int r  = (tid >> 6) + 4 * i;     // 0..31
            const int dt = r >> 4;
            const int hh = r & 15;
            const size_t gidx =
                (((size_t)bc * T + (size_t)(2 * t + dt)) * H
                 + (size_t)(16 * hb + hh)) * (size_t)W + (size_t)(4 * c);
            const uint32_t ldsOff = smBase + (uint32_t)(r * 1024) + ldsCol;
            asm volatile(
                "global_load_async_to_lds_b128 %0, %1, off"
                :
                : "v"(ldsOff), "v"((uint64_t)(uintptr_t)(x + gidx))
                : "memory");
        }
        asm volatile("s_wait_asynccnt 0" ::: "memory");
    }
    __syncthreads();

    // ------------------------------------------------------------------
    // Stage 2: full 2x2x2 butterfly -> all 8 subbands per thread.
    // thread -> (output row k = tid>>5 [wave-uniform], cols 4*w4..4*w4+3).
    // ------------------------------------------------------------------
    const int k  = tid >> 5;             // output row within tile (0..7)
    const int w4 = tid & 31;             // group of 4 output columns

    const int e0 = lds_slot(2 * w4);     // slot of floats 8w4..8w4+3
    const int e1 = e0 ^ 1;               // slot of floats 8w4+4..8w4+7

    float f[2][2][8];                    // [dt][dh][input w offset]
#pragma unroll
    for (int dt = 0; dt < 2; ++dt) {
#pragma unroll
        for (int dh = 0; dh < 2; ++dh) {
            const int r = dt * 16 + 2 * k + dh;
            v4f a = sm[r * 64 + e0];
            v4f b = sm[r * 64 + e1];
            f[dt][dh][0] = a.x; f[dt][dh][1] = a.y;
            f[dt][dh][2] = a.z; f[dt][dh][3] = a.w;
            f[dt][dh][4] = b.x; f[dt][dh][5] = b.y;
            f[dt][dh][6] = b.z; f[dt][dh][7] = b.w;
        }
    }

    const float Rc = 0.70710678118654752440f;   // 1/sqrt(2)

    float gl[2][8], gh[2][8];            // temporal low/high per dh row
#pragma unroll
    for (int dh = 0; dh < 2; ++dh) {
#pragma unroll
        for (int w = 0; w < 8; ++w) {
            gl[dh][w] = Rc * (f[0][dh][w] + f[1][dh][w]);
            gh[dh][w] = Rc * (f[0][dh][w] - f[1][dh][w]);
        }
    }

    float v[4][8];                       // [2*tH + vH][w]
#pragma unroll
    for (int w = 0; w < 8; ++w) {
        v[0][w] = Rc * (gl[0][w] + gl[1][w]);   // t:L v:L
        v[1][w] = Rc * (gl[0][w] - gl[1][w]);   // t:L v:H
        v[2][w] = Rc * (gh[0][w] + gh[1][w]);   // t:H v:L
        v[3][w] = Rc * (gh[0][w] - gh[1][w]);   // t:H v:H
    }

    // horizontal stage; scale = RESCALE * r = 2
    v4f o[8];
#pragma unroll
    for (int tv = 0; tv < 4; ++tv) {
        v4f lo, hi;
        lo.x = 2.0f * (v[tv][0] + v[tv][1]);  hi.x = 2.0f * (v[tv][0] - v[tv][1]);
        lo.y = 2.0f * (v[tv][2] + v[tv][3]);  hi.y = 2.0f * (v[tv][2] - v[tv][3]);
        lo.z = 2.0f * (v[tv][4] + v[tv][5]);  hi.z = 2.0f * (v[tv][4] - v[tv][5]);
        lo.w = 2.0f * (v[tv][6] + v[tv][7]);  hi.w = 2.0f * (v[tv][6] - v[tv][7]);
        o[2 * tv]     = lo;                    // s = 4*tH + 2*vH + 0
        o[2 * tv + 1] = hi;                    // s = 4*tH + 2*vH + 1
    }

    const int b = bc / 3;
    const int c = bc - 3 * b;
    const int h = 8 * hb + k;
    // out index for subband s: ((b*24 + s*3 + c)*T2 + t)*H2*W2 + h*W2 + 4*w4
    const size_t chanStride = (size_t)3 * T2 * H2 * W2;       // per-subband hop
    size_t oidx =
        ((((size_t)b * (8 * Cc) + (size_t)c) * T2 + (size_t)t) * H2
         + (size_t)h) * (size_t)W2 + (size_t)(4 * w4);
#pragma unroll
    for (int s = 0; s < 8; ++s) {
        __builtin_nontemporal_store(o[s], (v4f*)(out + oidx));
        oidx += chanStride;
    }
}

extern "C" void kernel_launch(void* const* d_in, const int* in_sizes, int n_in,
                              void* d_out, int out_size, void* d_ws, size_t ws_size,
                              hipStream_t stream) {
    (void)in_sizes; (void)n_in; (void)d_ws; (void)ws_size; (void)out_size;
    const float* x = (const float*)d_in[0];
    float* out = (float*)d_out;
    // One block per 8 output rows: grid = (H2/8, T2, B*C) = (16, 16, 12)
    dim3 grid(H2 / 8, T2, 4 * Cc);
    dim3 block(256);
    haar3d_dwt_kernel<<<grid, block, 0, stream>>>(x, out);
}